// Encoder_84585085927968
// MI455X (gfx1250) — compile-verified
//
#include <hip/hip_runtime.h>
#include <hip/hip_bf16.h>

typedef __attribute__((ext_vector_type(2))) float v2f;
typedef __attribute__((ext_vector_type(8))) float v8f;

#define VOCAB   50000
#define RAW_DIM 300
#define D_MODEL 256
#define D_STATE 32
#define BATCH   32
#define SEQ     1024

// -----------------------------------------------------------------------------
// Kernel 1: fused embedding-gather + linear projection via V_WMMA_F32_16X16X4_F32
//   X[g][d] = sum_r emb[src[g]][r] * W[d][r] + b_lin[d], masked by lengths.
// Block = 128 threads = 4 waves. Block owns 16 token rows (M), each wave owns
// a 16x64 N-slice (4 accumulator tiles). 75 K-steps of 4 -> 300 WMMAs/wave.
// -----------------------------------------------------------------------------
__global__ __launch_bounds__(128)
void proj_wmma_kernel(const int* __restrict__ src,
                      const int* __restrict__ lengths,
                      const float* __restrict__ emb,
                      const float* __restrict__ W,
                      const float* __restrict__ b_lin,
                      float* __restrict__ X)
{
    const int lane  = threadIdx.x & 31;
    const int wave  = threadIdx.x >> 5;          // 0..3
    const int mTile = blockIdx.x;                // 16 tokens per tile
    const int nBase = wave * 64;

    const int mIdx = lane & 15;                  // row within tile (A) / col (B,C,D)
    const int kHi  = (lane >> 4) * 2;            // lanes 16-31 hold K+2,K+3

    const int gRow0 = mTile * 16;                // flat token index of row 0
    // 16 rows of a tile never cross a batch boundary (1024 % 16 == 0)
    const int bb   = gRow0 >> 10;
    const int len  = lengths[bb];
    const int lOff = gRow0 & (SEQ - 1);

    // A-matrix row pointer: gathered embedding row for this lane's M
    const int tok = src[gRow0 + mIdx];
    const float* __restrict__ arow = emb + (size_t)tok * RAW_DIM;

    // B-matrix row pointers: W is (D_MODEL, RAW_DIM), B[k][n] = W[n][k]
    const float* __restrict__ wrow0 = W + (size_t)(nBase + 0 * 16 + mIdx) * RAW_DIM;
    const float* __restrict__ wrow1 = W + (size_t)(nBase + 1 * 16 + mIdx) * RAW_DIM;
    const float* __restrict__ wrow2 = W + (size_t)(nBase + 2 * 16 + mIdx) * RAW_DIM;
    const float* __restrict__ wrow3 = W + (size_t)(nBase + 3 * 16 + mIdx) * RAW_DIM;

    v8f c0 = {}, c1 = {}, c2 = {}, c3 = {};

    for (int k = 0; k < RAW_DIM; k += 4) {
        const int kl = k + kHi;                  // even; rows are 8B-aligned
        v2f a  = *(const v2f*)(arow  + kl);
        v2f b0 = *(const v2f*)(wrow0 + kl);
        v2f b1 = *(const v2f*)(wrow1 + kl);
        v2f b2 = *(const v2f*)(wrow2 + kl);
        v2f b3 = *(const v2f*)(wrow3 + kl);
        c0 = __builtin_amdgcn_wmma_f32_16x16x4_f32(false, a, false, b0, (short)0, c0, false, false);
        c1 = __builtin_amdgcn_wmma_f32_16x16x4_f32(false, a, false, b1, (short)0, c1, false, false);
        c2 = __builtin_amdgcn_wmma_f32_16x16x4_f32(false, a, false, b2, (short)0, c2, false, false);
        c3 = __builtin_amdgcn_wmma_f32_16x16x4_f32(false, a, false, b3, (short)0, c3, false, false);
    }

    // Bias per output column of each sub-tile
    const float bl0 = b_lin[nBase + 0 * 16 + mIdx];
    const float bl1 = b_lin[nBase + 1 * 16 + mIdx];
    const float bl2 = b_lin[nBase + 2 * 16 + mIdx];
    const float bl3 = b_lin[nBase + 3 * 16 + mIdx];

    // C/D layout: VGPR j, lanes 0-15 -> M=j, lanes 16-31 -> M=j+8, N = lane&15
    const int mHalf = (lane >> 4) * 8;
#pragma unroll
    for (int j = 0; j < 8; ++j) {
        const int m = j + mHalf;
        const int g = gRow0 + m;
        const bool valid = (lOff + m) < len;     // mask: x *= (l < length)
        float* __restrict__ orow = X + (size_t)g * D_MODEL + nBase + mIdx;
        orow[0 * 16] = valid ? (c0[j] + bl0) : 0.0f;
        orow[1 * 16] = valid ? (c1[j] + bl1) : 0.0f;
        orow[2 * 16] = valid ? (c2[j] + bl2) : 0.0f;
        orow[3 * 16] = valid ? (c3[j] + bl3) : 0.0f;
    }
}

// -----------------------------------------------------------------------------
// Kernel 2: diagonal SSM scan (ZOH). One lane per (b,d) channel; all 32 states
// and parameters live in registers. Per timestep: 1 coalesced load, 64 FMAs + 32 muls
// across 32 independent chains, 1 coalesced store.
// -----------------------------------------------------------------------------
__global__ __launch_bounds__(256)
void ssm_scan_kernel(const float* __restrict__ X,
                     const float* __restrict__ A_log,
                     const float* __restrict__ log_dt,
                     const float* __restrict__ Bp,
                     const float* __restrict__ Cp,
                     const float* __restrict__ Dp,
                     float* __restrict__ out)
{
    const int tid = blockIdx.x * blockDim.x + threadIdx.x;   // 0..8191
    const int b = tid >> 8;          // batch
    const int d = tid & (D_MODEL - 1);

    const float dt = __expf(log_dt[d]);
    float abar[D_STATE], bbar[D_STATE], cp[D_STATE], h[D_STATE];
#pragma unroll
    for (int n = 0; n < D_STATE; ++n) {
        const float A = -__expf(A_log[d * D_STATE + n]);
        abar[n] = __expf(dt * A);                 // exp(dt*A)
        bbar[n] = dt * Bp[d * D_STATE + n];       // dt*B
        cp[n]   = Cp[d * D_STATE + n];
        h[n]    = 0.0f;
    }
    const float Dd = Dp[d];

    const float* __restrict__ xp = X   + ((size_t)b * SEQ) * D_MODEL + d;
    float* __restrict__       yp = out + ((size_t)b * SEQ) * D_MODEL + d;

    for (int t = 0; t < SEQ; ++t) {
        const float x = xp[(size_t)t * D_MODEL];
        float y = Dd * x;
#pragma unroll
        for (int n = 0; n < D_STATE; ++n) {
            h[n] = __builtin_fmaf(abar[n], h[n], bbar[n] * x);
            y    = __builtin_fmaf(h[n], cp[n], y);
        }
        yp[(size_t)t * D_MODEL] = y;
    }
}

extern "C" void kernel_launch(void* const* d_in, const int* in_sizes, int n_in,
                              void* d_out, int out_size, void* d_ws, size_t ws_size,
                              hipStream_t stream) {
    const int*   src     = (const int*)d_in[0];
    const int*   lengths = (const int*)d_in[1];
    const float* emb     = (const float*)d_in[2];
    const float* W_lin   = (const float*)d_in[3];
    const float* b_lin   = (const float*)d_in[4];
    const float* A_log   = (const float*)d_in[5];
    const float* log_dt  = (const float*)d_in[6];
    const float* Bp      = (const float*)d_in[7];
    const float* Cp      = (const float*)d_in[8];
    const float* Dp      = (const float*)d_in[9];

    float* X = (float*)d_ws;   // (B*SEQ, D_MODEL) fp32 = 33.5 MB scratch

    // GEMM: 2048 tiles of 16 tokens, 4 waves per block cover N=256
    proj_wmma_kernel<<<(BATCH * SEQ) / 16, 128, 0, stream>>>(
        src, lengths, emb, W_lin, b_lin, X);

    // Scan: one lane per (b,d) channel -> 8192 threads
    ssm_scan_kernel<<<(BATCH * D_MODEL) / 256, 256, 0, stream>>>(
        X, A_log, log_dt, Bp, Cp, Dp, (float*)d_out);
}